// Jitter_85727547228504
// MI455X (gfx1250) — compile-verified
//
#include <hip/hip_runtime.h>

// out[b, t, :] = x[b, indices[t], :]
// B=64, T=4096, C=256, fp32. Pure HBM-bandwidth-bound gather:
// 256 MiB in + 256 MiB out -> ~23 us floor at 23.3 TB/s.
//
// CDNA5 data-movement choices:
//  - one wave32 per 1024-byte row; each lane moves 2x float4
//    -> global_load_b128 x2 + global_store_b128 x2 per thread, fully coalesced
//  - wave-uniform gather index forced to SGPR via readfirstlane
//    -> scalar-base (SADDR) + per-lane VGPR offset addressing
//  - non-temporal stores (th:NT) for the write-once output stream so it
//    does not displace the input stream in the 192 MB L2

typedef float v4f __attribute__((ext_vector_type(4)));

constexpr unsigned Bn   = 64;
constexpr unsigned Tn   = 4096;   // power of two
constexpr unsigned Cn   = 256;    // floats per row (1024 bytes)
constexpr unsigned ROWS = Bn * Tn;
constexpr unsigned WAVES_PER_BLOCK = 8;     // 256 threads / wave32
constexpr unsigned BLOCK = 32 * WAVES_PER_BLOCK;

__global__ __launch_bounds__(BLOCK) void jitter_gather_kernel(
    const float* __restrict__ x,
    const int*   __restrict__ indices,
    float*       __restrict__ out) {
  const unsigned lane = threadIdx.x & 31u;
  const unsigned wid  = threadIdx.x >> 5u;
  const unsigned row  = blockIdx.x * WAVES_PER_BLOCK + wid;   // [0, ROWS)

  const unsigned t = row & (Tn - 1u);   // row % T
  const unsigned b = row >> 12u;        // row / T   (T = 4096)

  // idx[t] is uniform across the wave: pin it to an SGPR so the source
  // address is SGPR64 base + per-lane VGPR offset (GVS addressing).
  const int j = __builtin_amdgcn_readfirstlane(indices[t]);

  const v4f* __restrict__ src =
      (const v4f*)(x + ((size_t)b * Tn + (size_t)(unsigned)j) * Cn);
  v4f* __restrict__ dst = (v4f*)(out + (size_t)row * Cn);

  // 1024-byte row: 64 float4s, 32 lanes -> 2 per lane, 512 B apart.
  v4f v0 = src[lane];
  v4f v1 = src[lane + 32u];

  // Output is written once and never re-read by the kernel: non-temporal.
  __builtin_nontemporal_store(v0, dst + lane);
  __builtin_nontemporal_store(v1, dst + lane + 32u);
}

extern "C" void kernel_launch(void* const* d_in, const int* in_sizes, int n_in,
                              void* d_out, int out_size, void* d_ws, size_t ws_size,
                              hipStream_t stream) {
  const float* x       = (const float*)d_in[0];
  const int*   indices = (const int*)d_in[1];
  float*       out     = (float*)d_out;

  dim3 grid(ROWS / WAVES_PER_BLOCK);   // 32768 blocks
  dim3 block(BLOCK);                   // 256 threads = 8 waves
  jitter_gather_kernel<<<grid, block, 0, stream>>>(x, indices, out);
}